// DifferentialMaxtree_62732292325969
// MI455X (gfx1250) — compile-verified
//
#include <hip/hip_runtime.h>
#include <hip/hip_bf16.h>
#include <math.h>

// DifferentialMaxtree on MI455X (gfx1250):
//   phase 1: per-node gaussian score via WMMA Gram-diagonal (f32 16x16x4)
//   phase 2: 22 pointer-doubling rounds (ping-pong in d_ws), L2-resident gathers
//
// d_in order: parent(i32,N), diff(f32,N), attributes(f32,N*15),
//             mean(f32,17), inv_diagonal_cov(f32,17)
// d_out: f32, N = 2048*2048

#define NN   (2048 * 2048)
#define EPSF 1e-10f
#define ROUNDS 22  // ceil(log2(4194304)) == 22

typedef __attribute__((ext_vector_type(2))) float v2f;
typedef __attribute__((ext_vector_type(8))) float v8f;

// ---------------------------------------------------------------------------
// Phase 1: contrib[i] = diff[i] * exp(-sum_j w_j * (feat_ij - mean_j)^2)
//
// Each wave32 handles 16 nodes. Lanes l and l+16 both build the feature
// vector of node (l & 15); with that duplication the WMMA A-operand
// (16x4 f32: lanes 0-15 carry K=k0,k0+1; lanes 16-31 carry K=k0+2,k0+3)
// and the B-operand (4x16 = A^T) are the *same* lane-local registers, so the
// 16x16 Gram matrix G*G^T needs no cross-lane data movement. Five 16x16x4
// WMMAs cover the 17 features (padded to 20); s_i is the Gram diagonal.
// ---------------------------------------------------------------------------
__global__ __launch_bounds__(256) void mt_score_kernel(
    const float* __restrict__ diff,
    const float* __restrict__ attr,
    const float* __restrict__ mean,
    const float* __restrict__ icov,
    float* __restrict__ contrib)
{
    const int tid  = blockIdx.x * blockDim.x + threadIdx.x;
    const int wave = tid >> 5;               // global wave id == tile id
    const int lane = threadIdx.x & 31;
    const int node = wave * 16 + (lane & 15);
    if (node >= NN) return;

    // raw attributes (xmin,ymin,xmax,ymax,area,angle,pca_big,pca_small,hu1..7)
    float f[15];
    const float* ap = attr + (long)node * 15;
#pragma unroll
    for (int j = 0; j < 15; ++j) f[j] = ap[j];

    // 17 derived features, reference order:
    // [f0..f4, log|f6..f14|, lshape, cos(angle), sin(angle)]
    float feats[17];
    feats[0] = f[0]; feats[1] = f[1]; feats[2] = f[2];
    feats[3] = f[3]; feats[4] = f[4];
#pragma unroll
    for (int j = 0; j < 9; ++j)
        feats[5 + j] = __logf(fabsf(f[6 + j]) + EPSF);
    feats[14] = sqrtf(f[7]) / (sqrtf(f[6]) + EPSF);
    feats[15] = __cosf(f[5]);
    feats[16] = __sinf(f[5]);

    // g = sqrt(max(icov,0)) * (feats - mean), zero-padded to K=20
    float g[20];
#pragma unroll
    for (int j = 0; j < 17; ++j) {
        float w = icov[j];
        w = (w > 0.0f) ? w : 0.0f;
        g[j] = (feats[j] - mean[j]) * sqrtf(w);
    }
    g[17] = 0.0f; g[18] = 0.0f; g[19] = 0.0f;

    const bool hi = (lane >= 16);            // high half-wave carries K+2,K+3
    v8f acc = {};
#pragma unroll
    for (int c = 0; c < 5; ++c) {            // K chunks: features 4c..4c+3
        v2f op;
        op.x = hi ? g[4 * c + 2] : g[4 * c + 0];
        op.y = hi ? g[4 * c + 3] : g[4 * c + 1];
        // D = A*B + C with A = tile of G, B = A^T (same registers)
        acc = __builtin_amdgcn_wmma_f32_16x16x4_f32(
            /*neg_a=*/false, op, /*neg_b=*/false, op,
            /*c_mod=*/(short)0, acc, /*reuse_a=*/false, /*reuse_b=*/false);
    }

    // Gram diagonal D[m][m]:
    //   m in 0..7  -> VGPR m,   lane m
    //   m in 8..15 -> VGPR m-8, lane m+16
    const bool active = (lane < 8) || (lane >= 24);
    const int  sel    = (lane < 8) ? lane : (lane - 24);
    float s = acc[0];
#pragma unroll
    for (int r = 1; r < 8; ++r)
        if (sel == r) s = acc[r];            // compile-time vector indices

    if (active) {
        const int m   = (lane < 8) ? lane : (lane - 16);
        const int out = wave * 16 + m;
        contrib[out] = diff[out] * __expf(-s);
    }
}

// ---------------------------------------------------------------------------
// Phase 2: one pointer-doubling round.
//   out_val[i] = val[i] + (ptr[i]>=0 ? val[ptr[i]] : 0)
//   out_ptr[i] =          (ptr[i]>=0 ? ptr[ptr[i]] : -1)
// Ping-pong buffers: each round reads a consistent snapshot. 32 MB working
// set stays resident in the 192 MB L2; late rounds (ptr == -1 almost
// everywhere) skip the gathers and become pure streaming passes.
// ---------------------------------------------------------------------------
__global__ __launch_bounds__(256) void mt_double_kernel(
    const float* __restrict__ vin, const int* __restrict__ pin,
    float* __restrict__ vout, int* __restrict__ pout)
{
    const int i = blockIdx.x * blockDim.x + threadIdx.x;
    if (i >= NN) return;
    int   p = pin[i];
    float v = vin[i];
    if (p >= 0) {
        v += vin[p];
        p  = pin[p];
    }
    vout[i] = v;
    pout[i] = p;
}

extern "C" void kernel_launch(void* const* d_in, const int* in_sizes, int n_in,
                              void* d_out, int out_size, void* d_ws, size_t ws_size,
                              hipStream_t stream) {
    const int*   parent = (const int*)  d_in[0];
    const float* diff   = (const float*)d_in[1];
    const float* attr   = (const float*)d_in[2];
    const float* mean   = (const float*)d_in[3];
    const float* icov   = (const float*)d_in[4];
    float*       out    = (float*)d_out;

    // workspace layout: valA | valB | ptrA | ptrB  (64 MB total)
    float* valA = (float*)d_ws;
    float* valB = valA + NN;
    int*   ptrA = (int*)(valB + NN);
    int*   ptrB = ptrA + NN;

    // phase 1: 16 nodes per wave, 8 waves per block -> NN/128 blocks
    mt_score_kernel<<<NN / 128, 256, 0, stream>>>(diff, attr, mean, icov, valA);

    // phase 2: 22 rounds, ping-pong; final round writes straight to d_out
    const float* vin  = valA;
    const int*   pin  = parent;
    float*       vbuf = valB;
    int*         pbuf = ptrB;
    for (int r = 0; r < ROUNDS; ++r) {
        float* vdst = (r == ROUNDS - 1) ? out : vbuf;
        mt_double_kernel<<<NN / 256, 256, 0, stream>>>(vin, pin, vdst, pbuf);
        vin = vdst;
        pin = pbuf;
        vbuf = (vbuf == valB) ? valA : valB;
        pbuf = (pbuf == ptrB) ? ptrA : ptrB;
    }
}